// QueryBasedDecoder_37546604102322
// MI455X (gfx1250) — compile-verified
//
#include <hip/hip_runtime.h>

typedef __attribute__((ext_vector_type(2))) float v2f;
typedef __attribute__((ext_vector_type(4))) float v4f;
typedef __attribute__((ext_vector_type(8))) float v8f;

#define BB   32
#define NN   1024
#define DD   128
#define CC   16
#define PP   16    // P*P
#define HG   32    // sqrt(N)
#define WAVES_PER_BLOCK 8

__global__ __launch_bounds__(256)
void qbd_fused_kernel(const float* __restrict__ x,     // [B,N,D]
                      const float* __restrict__ mask,  // [B,C]
                      const float* __restrict__ emb,   // [256,D] (use first 16 rows)
                      const float* __restrict__ W,     // [16,D]
                      const float* __restrict__ bias,  // [16]
                      float* __restrict__ out)         // [B,C,128,128]
{
    __shared__ __align__(16) float s_xw[WAVES_PER_BLOCK][16][16];
    __shared__ __align__(16) float s_ew[WAVES_PER_BLOCK][16][16];

    const int lane = threadIdx.x & 31;
    const int wave = threadIdx.x >> 5;
    const int tile = blockIdx.x * WAVES_PER_BLOCK + wave;   // 2048 tiles of 16 rows
    const long rowBase = (long)tile * 16;                   // global row in [0, B*N)
    const int b  = (int)(rowBase >> 10);                    // row / 1024
    const int n0 = (int)(rowBase & 1023);
    const int h  = n0 >> 5;                                 // constant across the 16-row tile
    const int g0 = n0 & 31;

    const int mrow  = lane & 15;   // M row (A) / N col (B,C) within the 16x16 tile
    const int khalf = lane >> 4;   // which K pair this lane holds

    // A fragment sources: x rows (for xw) and emb rows (for ew); B fragment: W rows.
    const float* xA = x   + (rowBase + mrow) * DD + khalf * 2;
    const float* eA = emb + (long)mrow * DD + khalf * 2;
    const float* wB = W   + (long)mrow * DD + khalf * 2;

    // acc_e initialized to bias[p] (p == N == mrow) so bias folds in for free.
    const float bp = bias[mrow];
    v8f acc  = {0.f,0.f,0.f,0.f,0.f,0.f,0.f,0.f};
    v8f acce = {bp,bp,bp,bp,bp,bp,bp,bp};

    // 32 K-steps of V_WMMA_F32_16X16X4_F32; B fragment shared by both chains.
    for (int k = 0; k < DD; k += 4) {
        v2f a  = *(const v2f*)(xA + k);
        v2f ae = *(const v2f*)(eA + k);
        v2f bw = *(const v2f*)(wB + k);
        acc  = __builtin_amdgcn_wmma_f32_16x16x4_f32(false, a,  false, bw,
                                                     (short)0, acc,  false, false);
        acce = __builtin_amdgcn_wmma_f32_16x16x4_f32(false, ae, false, bw,
                                                     (short)0, acce, false, false);
    }

    // Spill both 16x16 results to this wave's LDS slice.
    // C/D layout: VGPR r, lane l -> (M = r + 8*khalf, N = mrow)
    #pragma unroll
    for (int r = 0; r < 8; ++r) {
        s_xw[wave][r + 8 * khalf][mrow] = acc[r];
        s_ew[wave][r + 8 * khalf][mrow] = acce[r];
    }
    // Same-wave LDS RAW: backend inserts s_wait_dscnt; memory clobber keeps order.
    asm volatile("" ::: "memory");

    // Write phase: out[b, c, h*4+pr, (g0+nl)*4 + 0..3] as coalesced b128 stores.
    const float* maskb = mask + b * CC;
    float* outb = out + (long)b * CC * 128 * 128;
    const int nl = lane & 15;

    #pragma unroll 1
    for (int c = 0; c < CC; ++c) {
        const float m = maskb[c];
        #pragma unroll
        for (int j = 0; j < 2; ++j) {
            const int pr = (lane >> 4) + 2 * j;       // 0..3 across lane halves / j
            v4f xv = *(const v4f*)&s_xw[wave][nl][pr * 4];
            v4f ev = *(const v4f*)&s_ew[wave][c][pr * 4];
            v4f val = (xv + ev) * m;
            float* dst = outb + ((long)c * 128 + h * 4 + pr) * 128 + (g0 + nl) * 4;
            *(v4f*)dst = val;
        }
    }
}

extern "C" void kernel_launch(void* const* d_in, const int* in_sizes, int n_in,
                              void* d_out, int out_size, void* d_ws, size_t ws_size,
                              hipStream_t stream) {
    const float* x    = (const float*)d_in[0];
    const float* mask = (const float*)d_in[1];
    const float* emb  = (const float*)d_in[2];
    const float* W    = (const float*)d_in[3];
    const float* bias = (const float*)d_in[4];
    float* out = (float*)d_out;

    // 2048 tiles / 8 waves per block = 256 blocks x 256 threads (8 wave32s).
    qbd_fused_kernel<<<256, 256, 0, stream>>>(x, mask, emb, W, bias, out);
}